// MultiPrototypeLoss_63900523429988
// MI455X (gfx1250) — compile-verified
//
#include <hip/hip_runtime.h>
#include <hip/hip_bf16.h>

typedef __attribute__((ext_vector_type(2))) float v2f;
typedef __attribute__((ext_vector_type(8))) float v8f;

#define MP_BATCH        16384
#define MP_NUM_CLASSES  1000
#define MP_P            4
#define MP_ROWS         16       // rows (batch samples) per workgroup
#define MP_THREADS      256      // 8 wave32s

// Kernel 1: per 16-row tile -> stream X (coalesced float4), soft-min per class
// into LDS, WMMA f32 16x16x4 row-sum over classes, emit per-row ratio.
__global__ __launch_bounds__(MP_THREADS) void mp_loss_phase1(
    const float* __restrict__ X,      // (BATCH, NUM_CLASSES*P) f32
    const int*   __restrict__ labels, // (BATCH,) int
    float*       __restrict__ ratios) // (BATCH,) f32 scratch
{
    __shared__ float lds_d[MP_ROWS * MP_NUM_CLASSES]; // 64000 B
    __shared__ float lds_in[MP_ROWS];
    __shared__ float lds_sum[MP_ROWS];

    const int row_base = blockIdx.x * MP_ROWS;
    const int tid = threadIdx.x;

    // ---- Phase 1: streaming soft-min over P=4 prototypes per class ----
    // Each lane loads one class group = 16 contiguous bytes; a wave covers
    // 512 contiguous bytes per issue -> fully coalesced b128 loads.
    for (int row = 0; row < MP_ROWS; ++row) {
        const float4* xr =
            (const float4*)(X + (size_t)(row_base + row) * (MP_NUM_CLASSES * MP_P));
        for (int c = tid; c < MP_NUM_CLASSES; c += MP_THREADS) {
            float4 x = xr[c];
            float m = fminf(fminf(x.x, x.y), fminf(x.z, x.w));
            float s = expf(m - x.x) + expf(m - x.y) +
                      expf(m - x.z) + expf(m - x.w);
            // d = -logsumexp(-x) = min(x) - log(sum exp(min - x))
            lds_d[row * MP_NUM_CLASSES + c] = m - logf(s);
        }
    }
    __syncthreads();

    // in-class score for each row (label gather from LDS tile)
    if (tid < MP_ROWS) {
        int lab = labels[row_base + tid];
        lds_in[tid] = lds_d[tid * MP_NUM_CLASSES + lab];
    }

    // ---- Phase 2: row-sum over 1000 classes via V_WMMA_F32_16X16X4_F32 ----
    // D(16x16) = A(16x4) * ones(4x16) + C : every column of D is the rowsum
    // of A; chain C across 250 chunks of K=4. Full f32 precision.
    // A layout (32-bit A 16x4): lanes 0-15 hold K=0,1; lanes 16-31 hold K=2,3.
    if (tid < 32) {
        const int lane = tid;
        const int r    = lane & 15;
        const int koff = (lane >> 4) << 1; // 0 or 2
        const float* arow = &lds_d[r * MP_NUM_CLASSES + koff];

        v2f b; b[0] = 1.0f; b[1] = 1.0f;
        v8f acc = {};
        for (int c = 0; c < MP_NUM_CLASSES; c += 4) {
            float2 a2 = *(const float2*)(arow + c); // 8B-aligned LDS read
            v2f a; a[0] = a2.x; a[1] = a2.y;
            acc = __builtin_amdgcn_wmma_f32_16x16x4_f32(
                /*neg_a=*/false, a, /*neg_b=*/false, b,
                /*c_mod=*/(short)0, acc, /*reuse_a=*/false, /*reuse_b=*/false);
        }
        // D column N=0: lane 0 holds rows 0..7 in VGPRs 0..7,
        //               lane 16 holds rows 8..15 in VGPRs 0..7.
        if (lane == 0) {
#pragma unroll
            for (int i = 0; i < 8; ++i) lds_sum[i] = acc[i];
        }
        if (lane == 16) {
#pragma unroll
            for (int i = 0; i < 8; ++i) lds_sum[8 + i] = acc[i];
        }
    }
    __syncthreads();

    // ---- Phase 3: per-row ratio ----
    if (tid < MP_ROWS) {
        float total = lds_sum[tid];
        float inc   = lds_in[tid];
        float om    = (total - inc) * (1.0f / (float)(MP_NUM_CLASSES - 1));
        ratios[row_base + tid] = (inc * inc) / (om * om);
    }
}

// Kernel 2: deterministic single-block reduction of the 16384 ratios.
__global__ __launch_bounds__(MP_THREADS) void mp_loss_reduce(
    const float* __restrict__ ratios, float* __restrict__ out)
{
    __shared__ float red[MP_THREADS];
    float s = 0.0f;
    for (int i = threadIdx.x; i < MP_BATCH; i += MP_THREADS)
        s += ratios[i];
    red[threadIdx.x] = s;
    __syncthreads();
    for (int off = MP_THREADS / 2; off > 0; off >>= 1) {
        if (threadIdx.x < off) red[threadIdx.x] += red[threadIdx.x + off];
        __syncthreads();
    }
    if (threadIdx.x == 0)
        out[0] = red[0] * (1.0f / (float)MP_BATCH);
}

extern "C" void kernel_launch(void* const* d_in, const int* in_sizes, int n_in,
                              void* d_out, int out_size, void* d_ws, size_t ws_size,
                              hipStream_t stream) {
    (void)in_sizes; (void)n_in; (void)out_size; (void)ws_size;
    const float* X      = (const float*)d_in[0];
    const int*   labels = (const int*)d_in[1];
    float* ratios = (float*)d_ws;   // 16384 floats = 64 KB scratch
    float* out    = (float*)d_out;

    mp_loss_phase1<<<MP_BATCH / MP_ROWS, MP_THREADS, 0, stream>>>(X, labels, ratios);
    mp_loss_reduce<<<1, MP_THREADS, 0, stream>>>(ratios, out);
}